// multiplicativeNoise_62474594287635
// MI455X (gfx1250) — compile-verified
//
#include <hip/hip_runtime.h>
#include <stdint.h>

// ---------------------------------------------------------------------------
// SDE RK4 (multiplicative noise) for MI455X / gfx1250.
//  - drift is exactly 0 in fp32 (see analysis), so each stage is the bilinear
//    map  k[b,i] = sum_{j,k} G[i,j,k] u[b,j] dw[b,k]
//  - implemented as f32 WMMA GEMM:  M=16 rows, N=i (64), K=p=(j,k) (4096)
//    A[b,p] = u[b,j]*dw[b,k] built on the fly, B[p,i] = G[i,j,k] pre-packed.
// ---------------------------------------------------------------------------

typedef float v2f __attribute__((ext_vector_type(2)));
typedef float v8f __attribute__((ext_vector_type(8)));
typedef int   v4i __attribute__((vector_size(16)));   // matches async-LDS builtin param

#define DEV __device__ __forceinline__
#define AS1 __attribute__((address_space(1)))
#define AS3 __attribute__((address_space(3)))

constexpr int DIMN     = 64;
constexpr int NSTEP    = 100;
constexpr int BATCH    = 16384;
constexpr int NTHREADS = 256;           // 8 waves (wave32)
constexpr int RT       = 2;             // 16-row tiles per wave
constexpr int ROWS     = 8 * RT * 16;   // 256 rows per block
constexpr int PITCH    = 68;            // padded LDS row pitch (floats): 16B-aligned, conflict-free
constexpr int NPS      = (DIMN * DIMN) / 4;     // 1024 K-slices of 4
constexpr int CHUNK_PS = 32;                    // K-slices per G chunk
constexpr int NCHUNK   = NPS / CHUNK_PS;        // 32 chunks per increment
constexpr int CHUNK_FLOATS = CHUNK_PS * 4 * 64; // 8192 floats = 32KB
constexpr int GP_FLOATS    = NPS * 4 * 64;      // 262144 floats = 1MB packed G

#if defined(__HIP_DEVICE_COMPILE__) && \
    __has_builtin(__builtin_amdgcn_global_load_async_to_lds_b128) && \
    __has_builtin(__builtin_amdgcn_s_wait_asynccnt)
#define USE_ASYNC_LDS 1
#else
#define USE_ASYNC_LDS 0
#endif

#if USE_ASYNC_LDS
// per thread per chunk: 8 async b128 ops (8*16B*256thr = 32KB)
#define WAIT_ASYNC_PREV() __builtin_amdgcn_s_wait_asynccnt(8)
#define WAIT_ASYNC_ALL()  __builtin_amdgcn_s_wait_asynccnt(0)
#else
#define WAIT_ASYNC_PREV()
#define WAIT_ASYNC_ALL()
#endif

// Stage one 32KB chunk of packed G into LDS (cooperative, 256 threads).
DEV void stage_chunk(const float* __restrict__ gp, float* dst, int tid) {
#if USE_ASYNC_LDS
  const char* g = (const char*)gp + tid * 128;
  char*       l = (char*)dst + tid * 128;
#pragma unroll
  for (int q = 0; q < 8; ++q)
    __builtin_amdgcn_global_load_async_to_lds_b128(
        (AS1 v4i*)(g + q * 16), (AS3 v4i*)(l + q * 16), 0, 0);
#else
  const float4* g4 = (const float4*)gp + tid * 8;
  float4*       l4 = (float4*)dst + tid * 8;
#pragma unroll
  for (int q = 0; q < 8; ++q) l4[q] = g4[q];
#endif
}

// C[rt][nt] = bilinear(u, dw) over K = 4096 via V_WMMA_F32_16X16X4_F32.
DEV void bilinear(v8f (&C)[RT][4], const float* uS, const float* dwS, float* gS,
                  const float* __restrict__ Gp, int tid, int wrow, int lane,
                  int halv, int lr) {
  const v8f zero = {0.f, 0.f, 0.f, 0.f, 0.f, 0.f, 0.f, 0.f};
#pragma unroll
  for (int rt = 0; rt < RT; ++rt)
#pragma unroll
    for (int nt = 0; nt < 4; ++nt) C[rt][nt] = zero;

  stage_chunk(Gp, gS, tid);  // prologue: chunk 0 -> buffer 0
  for (int c = 0; c < NCHUNK; ++c) {
    if (c + 1 < NCHUNK) {
      stage_chunk(Gp + (size_t)(c + 1) * CHUNK_FLOATS,
                  gS + ((c + 1) & 1) * CHUNK_FLOATS, tid);
      WAIT_ASYNC_PREV();   // chunk c complete (async loads retire in order)
    } else {
      WAIT_ASYNC_ALL();
    }
    __syncthreads();
    const float* gb = gS + (c & 1) * CHUNK_FLOATS;
#pragma unroll
    for (int jj = 0; jj < 2; ++jj) {
      const int j = c * 2 + jj;
      float uv[RT];
#pragma unroll
      for (int rt = 0; rt < RT; ++rt)
        uv[rt] = uS[(wrow + rt * 16 + lr) * PITCH + j];
#pragma unroll
      for (int kk = 0; kk < 16; ++kk) {
        v2f bf[4];
#pragma unroll
        for (int nt = 0; nt < 4; ++nt)
          bf[nt] = *(const v2f*)(gb + ((jj * 16 + kk) * 4 + nt) * 64 + lane * 2);
#pragma unroll
        for (int rt = 0; rt < RT; ++rt) {
          v2f dwv = *(const v2f*)(dwS + (wrow + rt * 16 + lr) * PITCH +
                                  kk * 4 + 2 * halv);
          v2f a;
          a.x = uv[rt] * dwv.x;
          a.y = uv[rt] * dwv.y;
#pragma unroll
          for (int nt = 0; nt < 4; ++nt)
            C[rt][nt] = __builtin_amdgcn_wmma_f32_16x16x4_f32(
                false, a, false, bf[nt], (short)0, C[rt][nt], false, false);
        }
      }
    }
    __syncthreads();
  }
}

// uS[b,i] = Y + a*C   (fragment layout -> row-major LDS, disjoint rows/wave)
DEV void store_u(float* uS, const v8f (&Y)[RT][4], const v8f (&C)[RT][4],
                 float a, int wrow, int halv, int lr) {
#pragma unroll
  for (int rt = 0; rt < RT; ++rt)
#pragma unroll
    for (int v = 0; v < 8; ++v) {
      const int b = wrow + rt * 16 + v + halv * 8;
      float* dst = uS + b * PITCH + lr;
#pragma unroll
      for (int nt = 0; nt < 4; ++nt)
        dst[nt * 16] = Y[rt][nt][v] + a * C[rt][nt][v];
    }
}

// ---------------------------------------------------------------------------
// Kernel 1: pack G[i,j,k] (i*4096 + j*64 + k) into the exact B-fragment
// layout of V_WMMA_F32_16X16X4_F32: Gp[((ps*4+nt)*32+lane)*2+q] =
//   B[p = ps*4 + 2*(lane>>4) + q, i = nt*16 + (lane&15)] = G[i*4096 + p]
// ---------------------------------------------------------------------------
__global__ void pack_g_kernel(const float* __restrict__ G,
                              float* __restrict__ Gp) {
  int tid = blockIdx.x * blockDim.x + threadIdx.x;
  if (tid >= NPS * 4 * 32) return;
  int lane = tid & 31;
  int nt   = (tid >> 5) & 3;
  int ps   = tid >> 7;
  int halv = lane >> 4, lr = lane & 15;
  int i  = nt * 16 + lr;
  int p0 = ps * 4 + halv * 2;
  Gp[2 * tid + 0] = G[i * 4096 + p0 + 0];
  Gp[2 * tid + 1] = G[i * 4096 + p0 + 1];
}

// ---------------------------------------------------------------------------
// Kernel 2: 100-step RK4, 256 batch rows per block, output gather at idx[b].
// ---------------------------------------------------------------------------
__global__ __launch_bounds__(NTHREADS, 1)
void sde_rk4_kernel(const float* __restrict__ y0, const float* __restrict__ t,
                    const float* __restrict__ dW, const float* __restrict__ Gp,
                    float* __restrict__ out) {
  extern __shared__ float lds[];
  float* uS   = lds;                          // ROWS*PITCH
  float* dwS  = uS + ROWS * PITCH;            // ROWS*PITCH
  float* gS   = dwS + ROWS * PITCH;           // 2*CHUNK_FLOATS (double buffer)
  int*   idxS = (int*)(gS + 2 * CHUNK_FLOATS);// ROWS

  const int tid  = threadIdx.x;
  const int wave = tid >> 5;
  const int lane = tid & 31;
  const int halv = lane >> 4;
  const int lr   = lane & 15;
  const int wrow = wave * (RT * 16);
  const int rowBlock = blockIdx.x * ROWS;

  // per-row output step index: clamp(trunc(100*t), 0, 99)
  for (int r = tid; r < ROWS; r += NTHREADS) {
    float tv = t[rowBlock + r];
    int ix = (int)truncf(tv * (float)NSTEP);
    ix = ix < 0 ? 0 : (ix > NSTEP - 1 ? NSTEP - 1 : ix);
    idxS[r] = ix;
  }

  // y state in C-fragment layout: yv[rt][nt][v] = y[row = wrow+rt*16+v+8*halv,
  //                                                 col = nt*16+lr]
  v8f yv[RT][4];
#pragma unroll
  for (int rt = 0; rt < RT; ++rt)
#pragma unroll
    for (int nt = 0; nt < 4; ++nt)
#pragma unroll
      for (int v = 0; v < 8; ++v) {
        int b = wrow + rt * 16 + v + halv * 8;
        yv[rt][nt][v] = y0[(size_t)(rowBlock + b) * DIMN + nt * 16 + lr];
      }

  __syncthreads();

  for (int s = 0; s < NSTEP; ++s) {
    // stage dW[s] tile -> LDS (coalesced float4), and hint-prefetch next step
    {
      const float4* src =
          (const float4*)(dW + ((size_t)s * BATCH + rowBlock) * DIMN);
#pragma unroll 4
      for (int q = tid; q < ROWS * 16; q += NTHREADS) {
        int r = q >> 4, c4 = q & 15;
        *(float4*)(dwS + r * PITCH + c4 * 4) = src[q];
      }
      if (s + 1 < NSTEP) {
        const char* nxt =
            (const char*)(dW + ((size_t)(s + 1) * BATCH + rowBlock) * DIMN);
        __builtin_prefetch(nxt + tid * 256, 0, 0);
        __builtin_prefetch(nxt + tid * 256 + 128, 0, 0);
      }
    }

    v8f S[RT][4], C[RT][4];

    // stage 1: u1 = y
    store_u(uS, yv, yv, 0.0f, wrow, halv, lr);
    bilinear(C, uS, dwS, gS, Gp, tid, wrow, lane, halv, lr);  // k1
#pragma unroll
    for (int rt = 0; rt < RT; ++rt)
#pragma unroll
      for (int nt = 0; nt < 4; ++nt) S[rt][nt] = C[rt][nt];

    // stage 2: u2 = y + 0.5 k1
    store_u(uS, yv, C, 0.5f, wrow, halv, lr);
    bilinear(C, uS, dwS, gS, Gp, tid, wrow, lane, halv, lr);  // k2
#pragma unroll
    for (int rt = 0; rt < RT; ++rt)
#pragma unroll
      for (int nt = 0; nt < 4; ++nt) S[rt][nt] = S[rt][nt] + 2.0f * C[rt][nt];

    // stage 3: u3 = y + 0.5 k2
    store_u(uS, yv, C, 0.5f, wrow, halv, lr);
    bilinear(C, uS, dwS, gS, Gp, tid, wrow, lane, halv, lr);  // k3
#pragma unroll
    for (int rt = 0; rt < RT; ++rt)
#pragma unroll
      for (int nt = 0; nt < 4; ++nt) S[rt][nt] = S[rt][nt] + 2.0f * C[rt][nt];

    // stage 4: u4 = y + k3
    store_u(uS, yv, C, 1.0f, wrow, halv, lr);
    bilinear(C, uS, dwS, gS, Gp, tid, wrow, lane, halv, lr);  // k4

    // y += (k1 + 2k2 + 2k3 + k4) / 6
#pragma unroll
    for (int rt = 0; rt < RT; ++rt)
#pragma unroll
      for (int nt = 0; nt < 4; ++nt)
        yv[rt][nt] = yv[rt][nt] + (S[rt][nt] + C[rt][nt]) / 6.0f;

    // emit rows whose sample time lands on this step
#pragma unroll
    for (int rt = 0; rt < RT; ++rt)
#pragma unroll
      for (int v = 0; v < 8; ++v) {
        const int bl = wrow + rt * 16 + v + halv * 8;
        if (idxS[bl] == s) {
          float* o = out + (size_t)(rowBlock + bl) * DIMN + lr;
#pragma unroll
          for (int nt = 0; nt < 4; ++nt) o[nt * 16] = yv[rt][nt][v];
        }
      }
  }
}

// ---------------------------------------------------------------------------
extern "C" void kernel_launch(void* const* d_in, const int* in_sizes, int n_in,
                              void* d_out, int out_size, void* d_ws,
                              size_t ws_size, hipStream_t stream) {
  (void)in_sizes; (void)n_in; (void)out_size; (void)ws_size;
  const float* y0 = (const float*)d_in[0];
  const float* t  = (const float*)d_in[1];
  const float* G  = (const float*)d_in[2];
  const float* dW = (const float*)d_in[3];
  float* out = (float*)d_out;
  float* Gp  = (float*)d_ws;  // 1MB packed G

  pack_g_kernel<<<(NPS * 4 * 32 + 255) / 256, 256, 0, stream>>>(G, Gp);

  const size_t shmem = (size_t)(2 * ROWS * PITCH + 2 * CHUNK_FLOATS) *
                           sizeof(float) +
                       (size_t)ROWS * sizeof(int);  // ~206 KB of the 320 KB WGP LDS
  (void)hipFuncSetAttribute((const void*)sde_rk4_kernel,
                            hipFuncAttributeMaxDynamicSharedMemorySize,
                            (int)shmem);
  sde_rk4_kernel<<<BATCH / ROWS, NTHREADS, shmem, stream>>>(y0, t, dW, Gp, out);
}